// VADetector_3942779978171
// MI455X (gfx1250) — compile-verified
//
#include <hip/hip_runtime.h>
#include <math.h>

#define N_STATES 256
#define ML 8
#define LOG_SQRT_2PI 0.9189385332046727f   // log(sqrt(2*pi))

typedef float v2f __attribute__((ext_vector_type(2)));
typedef float v8f __attribute__((ext_vector_type(8)));

// ---------------------------------------------------------------------------
// Kernel 1: confident_bits / confidence_word via WMMA-generated priors.
//
// priors[t,s] = C - 0.5*(rx_t - sp_s)^2
//             = rx_t*sp_s + (-0.5*rx_t^2)*1 + 1*(C - 0.5*sp_s^2) + 0
// K=4 GEMM:  A[t,:] = [rx_t, -0.5 rx_t^2, 1, 0]
//            B[:,s] = [sp_s, 1, C - 0.5 sp_s^2, 0]^T
// One V_WMMA_F32_16X16X4_F32 per 16x16 (t,s) tile.
//
// A 16x4 f32 (ISA 7.12.2): lanes 0-15 -> M=lane, vgpr0=K0, vgpr1=K1;
//                          lanes 16-31 -> M=lane-16, vgpr0=K2, vgpr1=K3.
// B 4x16: rows striped across lanes: vgpr0 = [K0 row | K1 row],
//                                    vgpr1 = [K2 row | K3 row].
// D 16x16 f32: vgpr r -> row (r + 8*(lane>=16)), col = lane&15.
// ---------------------------------------------------------------------------
#define K1_WAVES   2
#define K1_ROWS    (K1_WAVES * 16)     // t-rows per block
#define K1_THREADS (K1_WAVES * 32)

__global__ __launch_bounds__(K1_THREADS)
void va_confidence_kernel(const float* __restrict__ rx,
                          const float* __restrict__ h,
                          float* __restrict__ out_bits,   // confident_bits
                          float* __restrict__ out_conf,   // confidence_word
                          int T) {
  __shared__ float sp_lds[N_STATES];
  __shared__ float pri[K1_ROWS][N_STATES];   // 32 KB

  const int tid  = threadIdx.x;
  const int lane = tid & 31;
  const int wave = tid >> 5;
  const bool lo  = (lane < 16);

  // state priors sp[s] = sum_j (1-2*bit_j(s)) * h[j], bits MSB-first
  for (int s = tid; s < N_STATES; s += K1_THREADS) {
    float acc = 0.f;
#pragma unroll
    for (int j = 0; j < ML; ++j) {
      int bit = (s >> (ML - 1 - j)) & 1;
      acc += (1.f - 2.f * (float)bit) * h[j];
    }
    sp_lds[s] = acc;
  }
  __syncthreads();

  const int t0   = blockIdx.x * K1_ROWS + wave * 16;
  const int tA   = t0 + (lane & 15);
  const float rxv = (tA < T) ? rx[tA] : 0.f;

  v2f a;
  a.x = lo ? rxv                : 1.0f;   // K0 | K2
  a.y = lo ? -0.5f * rxv * rxv  : 0.0f;   // K1 | K3

  // Preload all 16 sp values for this lane's column, unconditionally
  // (no EXEC-branched ds_load between WMMAs).
  float spn[16];
#pragma unroll
  for (int i = 0; i < 16; ++i)
    spn[i] = sp_lds[i * 16 + (lane & 15)];

  // Pre-built B fragments: pure v_cndmask selects, no branches.
  v2f bfr[16];
#pragma unroll
  for (int i = 0; i < 16; ++i) {
    bfr[i].x = lo ? spn[i]                                    : 1.0f; // K0|K1
    bfr[i].y = lo ? (LOG_SQRT_2PI - 0.5f * spn[i] * spn[i])   : 0.0f; // K2|K3
  }

  const int row_base = wave * 16 + (lo ? 0 : 8);
  const int col_base = lane & 15;
#pragma unroll
  for (int i = 0; i < 16; ++i) {
    v8f c = {};
    c = __builtin_amdgcn_wmma_f32_16x16x4_f32(
            /*neg_a=*/false, a, /*neg_b=*/false, bfr[i],
            /*c_mod=*/(short)0, c, /*reuse_a=*/false, /*reuse_b=*/false);
    const int col = i * 16 + col_base;
#pragma unroll
    for (int r = 0; r < 8; ++r)
      pri[row_base + r][col] = c[r];
  }
  __syncthreads();

  // per-t first-occurrence argmax of exp(priors) (strict > over ascending s)
  for (int m = tid; m < K1_ROWS; m += K1_THREADS) {
    const int t = blockIdx.x * K1_ROWS + m;
    if (t >= T) continue;
    float best = expf(pri[m][0]);
    int   bidx = 0;
    for (int s = 1; s < N_STATES; ++s) {
      float e = expf(pri[m][s]);    // each thread reads its own row m, stride 256 floats
      if (e > best) { best = e; bidx = s; }
    }
    out_bits[t] = (float)(bidx & 1);
    out_conf[t] = best;
  }
}

// ---------------------------------------------------------------------------
// Kernel 2: serial Viterbi min-plus scan, one wave32 wave.
//
// Layout: lane l, slot k holds state S = 8*(l%16) + k + 128*(l>=16).
// in_prob has exact period 128, so lanes 16-31 are bit-identical duplicates
// (indices +128) of lanes 0-15 -- needed because branch metrics differ
// across the full 256 prev states.
//
// Butterfly out[s'] = min(v[2s'], v[2s'+1]) pairs ADJACENT states -> purely
// within-lane (4 v_min_f32). Lane l emits out-states 4*(l%16)+j + 64*(l>=16).
// Redistribution = 8 ds_bpermute from constant lane 2*(l&15) + (k>>2).
//
// bit = argmin(in_prob) % 2 (pre-update). Duplicate half means the xor-16
// butterfly stage can never change the winner on lanes 0-15: only stages
// 8/4/2/1 are needed, and lane 0 ends with the first-occurrence argmin over
// states 0..127, which equals jnp.argmin over all 256 by periodicity.
// ---------------------------------------------------------------------------
__global__ __launch_bounds__(32)
void va_scan_kernel(const float* __restrict__ rx,
                    const float* __restrict__ h,
                    float* __restrict__ out_bits,
                    int T) {
  const int lane = threadIdx.x & 31;
  const int lam  = lane & 15;
  const int hi   = (lane >> 4) & 1;

  float hreg[ML];
#pragma unroll
  for (int j = 0; j < ML; ++j) hreg[j] = h[j];

  float spv[8];
  int   sidx[8];
#pragma unroll
  for (int k = 0; k < 8; ++k) {
    const int s = 8 * lam + k + 128 * hi;
    sidx[k] = s;
    float acc = 0.f;
#pragma unroll
    for (int j = 0; j < ML; ++j) {
      int bit = (s >> (ML - 1 - j)) & 1;
      acc += (1.f - 2.f * (float)bit) * hreg[j];
    }
    spv[k] = acc;
  }

  // target (lane, slot k) pulls out-state 8*lam+k: source lane simplifies to
  // 2*(l&15) + (k>>2) for ALL lanes (the &15 / +16 terms cancel), slot k&3.
  int srcLane[8];
#pragma unroll
  for (int k = 0; k < 8; ++k)
    srcLane[k] = 2 * lam + (k >> 2);

  float ip[8];
#pragma unroll
  for (int k = 0; k < 8; ++k) ip[k] = 0.f;

  float rxt = rx[0];
#pragma unroll 2
  for (int t = 0; t < T; ++t) {
    const int tn = (t + 1 < T) ? (t + 1) : (T - 1);   // uniform, SALU
    const float rxn = rx[tn];                          // prefetch next

    // --- bit decision BEFORE update ---------------------------------------
    // In-lane argmin as a depth-3 tree. At every node the right operand's
    // state index strictly exceeds the left's, so strict < keeps the
    // first-occurrence (smallest-index) winner.
    float v01 = ip[0]; int i01 = sidx[0];
    if (ip[1] < v01) { v01 = ip[1]; i01 = sidx[1]; }
    float v23 = ip[2]; int i23 = sidx[2];
    if (ip[3] < v23) { v23 = ip[3]; i23 = sidx[3]; }
    float v45 = ip[4]; int i45 = sidx[4];
    if (ip[5] < v45) { v45 = ip[5]; i45 = sidx[5]; }
    float v67 = ip[6]; int i67 = sidx[6];
    if (ip[7] < v67) { v67 = ip[7]; i67 = sidx[7]; }
    if (v23 < v01) { v01 = v23; i01 = i23; }
    if (v67 < v45) { v45 = v67; i45 = i67; }
    float bv = v01; int bi = i01;
    if (v45 < bv) { bv = v45; bi = i45; }

    // Cross-lane butterfly over the 16 distinct lanes only (xor-16 stage
    // provably redundant due to the duplicate half).
#pragma unroll
    for (int off = 8; off >= 1; off >>= 1) {
      float ov = __shfl_xor(bv, off, 32);
      int   oi = __shfl_xor(bi, off, 32);
      if (ov < bv || (ov == bv && oi < bi)) { bv = ov; bi = oi; }
    }
    if (lane == 0) out_bits[t] = (float)(bi & 1);

    // --- branch metrics + ACS: v = in_prob - priors[t] --------------------
    float v[8];
#pragma unroll
    for (int k = 0; k < 8; ++k) {
      const float d = rxt - spv[k];
      const float p = -(d * d * 0.5f - LOG_SQRT_2PI);
      v[k] = ip[k] - p;
    }
    float mm[4];
    mm[0] = fminf(v[0], v[1]);
    mm[1] = fminf(v[2], v[3]);
    mm[2] = fminf(v[4], v[5]);
    mm[3] = fminf(v[6], v[7]);

#pragma unroll
    for (int k = 0; k < 8; ++k)
      ip[k] = __shfl(mm[k & 3], srcLane[k], 32);

    rxt = rxn;
  }
}

// ---------------------------------------------------------------------------
extern "C" void kernel_launch(void* const* d_in, const int* in_sizes, int n_in,
                              void* d_out, int out_size, void* d_ws, size_t ws_size,
                              hipStream_t stream) {
  const float* rx = (const float*)d_in[0];
  // d_in[1] = phase (int scalar), unused by the math
  const float* h  = (const float*)d_in[2];
  float* out = (float*)d_out;
  const int T = in_sizes[0];

  // d_out layout: [detected_word(T) | confident_bits(T) | confidence_word(T)]
  const int nblk = (T + K1_ROWS - 1) / K1_ROWS;
  va_confidence_kernel<<<nblk, K1_THREADS, 0, stream>>>(rx, h, out + T, out + 2 * T, T);
  va_scan_kernel<<<1, 32, 0, stream>>>(rx, h, out, T);
}